// DefinableVectorRouting_30623116821357
// MI455X (gfx1250) — compile-verified
//
#include <hip/hip_runtime.h>

typedef __attribute__((ext_vector_type(2))) float v2f;
typedef __attribute__((ext_vector_type(8))) float v8f;

#define B_SZ 32
#define NI   1024
#define DI   1024
#define NO   64
#define DO   1024

__device__ __forceinline__ v8f wmma_f32(v2f a, v2f b, v8f c) {
  // D = A(16x4) x B(4x16) + C(16x16), all f32, wave32
  return __builtin_amdgcn_wmma_f32_16x16x4_f32(false, a, false, b, (short)0, c,
                                               false, false);
}

// ---------------------------------------------------------------------------
// Stage 1: f_a[b,i] = sigmoid(x[b,i,:] . W_A + b_A), one wave per row
// ---------------------------------------------------------------------------
__global__ void fa_kernel(const float* __restrict__ x, const float* __restrict__ WA,
                          const float* __restrict__ bA, float* __restrict__ fa) {
  const int wave = (blockIdx.x * blockDim.x + threadIdx.x) >> 5; // row = b*NI+i
  const int lane = threadIdx.x & 31;
  const float* row = x + (size_t)wave * DI;
  float s = 0.0f;
  #pragma unroll 4
  for (int d = lane; d < DI; d += 32) s += row[d] * WA[d];
  for (int off = 16; off > 0; off >>= 1) s += __shfl_xor(s, off, 32);
  if (lane == 0) fa[wave] = 1.0f / (1.0f + __expf(-(s + bA[0])));
}

// ---------------------------------------------------------------------------
// Stage 2a: iter-0 phi, stored transposed: phiT[(j*NI + i)*32 + b]
// phi = f_a[b,i] * (beta_use/NO - beta_ign*(1 - 1/NO))
// ---------------------------------------------------------------------------
__global__ void phi0_kernel(const float* __restrict__ fa, const float* __restrict__ bu,
                            const float* __restrict__ bi, float* __restrict__ phiT) {
  const int idx = blockIdx.x * blockDim.x + threadIdx.x; // = (j*NI + i)*32 + b
  const int b = idx & (B_SZ - 1);
  const int t = idx >> 5;          // j*NI + i
  const int i = t & (NI - 1);
  const int j = t >> 10;
  const float f = fa[b * NI + i];
  const float w = bu[i * NO + j] * (1.0f / NO) - bi[i * NO + j] * (1.0f - 1.0f / NO);
  phiT[idx] = f * w;               // contiguous writes
}

// ---------------------------------------------------------------------------
// phi (.) W_mem contraction:  out[b, j, h] = sum_i phiT[j][i][b] * Wmem[i, j, h]
// 64 GEMMs (one per j): M=32, K=1024, N=1024.
// Block: 256 thr (8 waves). A-panel phiT[j] (128 KB) staged in LDS, swizzled
// [k/4][b][k%4] so each lane's (k,k+1) pair is one ds_load_b64.
// Each wave owns two 16-wide N tiles (full M=32): per K-quad ->
// 2 ds_load_b64 + 4 global_load_b32 (W_mem stream) + 4 v_wmma.
// grid = (4, 64 j)
// ---------------------------------------------------------------------------
__global__ void memgemm_kernel(const float* __restrict__ phiT,
                               const float* __restrict__ Wmem,
                               float* __restrict__ out) {
  extern __shared__ float As[];    // NI*B_SZ floats = 128 KB
  const int j = blockIdx.y;

  // cooperative, coalesced fill of the swizzled A panel
  const float* Apanel = phiT + (size_t)j * NI * B_SZ;   // [k][b]
  for (int t = threadIdx.x; t < NI * B_SZ; t += 256) {
    const int k = t >> 5;
    const int b = t & 31;
    As[((k >> 2) << 7) + (b << 2) + (k & 3)] = Apanel[t];
  }
  __syncthreads();

  const int lane = threadIdx.x & 31;
  const int half = lane >> 4;
  const int lm   = lane & 15;
  const int wave = threadIdx.x >> 5;
  const int n0 = (blockIdx.x * 8 + wave) * 32;          // two tiles: n0, n0+16

  const v2f* AsP = (const v2f*)As;                      // pair idx: q*64 + b*2 + half
  const float* B0 = Wmem + (size_t)j * DO + n0 + lm;    // + k*(NO*DO)
  const float* B1 = B0 + 16;

  v8f acc00 = {}, acc01 = {}, acc10 = {}, acc11 = {};
  #pragma unroll 4
  for (int q = 0; q < NI / 4; ++q) {
    const v2f a0 = AsP[q * 64 + lm * 2 + half];         // batch rows 0..15
    const v2f a1 = AsP[q * 64 + (lm + 16) * 2 + half];  // batch rows 16..31
    const int kx = 4 * q + 2 * half;
    const int ky = kx + 1;
    v2f b0, b1;
    b0.x = B0[(size_t)kx * (NO * DO)];  b0.y = B0[(size_t)ky * (NO * DO)];
    b1.x = B1[(size_t)kx * (NO * DO)];  b1.y = B1[(size_t)ky * (NO * DO)];
    acc00 = wmma_f32(a0, b0, acc00);
    acc10 = wmma_f32(a1, b0, acc10);
    acc01 = wmma_f32(a0, b1, acc01);
    acc11 = wmma_f32(a1, b1, acc11);
  }
  #pragma unroll
  for (int v = 0; v < 8; ++v) {
    const int blo = v + 8 * half;      // batch rows 0..15
    const int bhi = blo + 16;          // batch rows 16..31
    float* o0 = out + ((size_t)blo * NO + j) * DO + n0 + lm;
    float* o1 = out + ((size_t)bhi * NO + j) * DO + n0 + lm;
    o0[0]  = acc00[v];
    o0[16] = acc01[v];
    o1[0]  = acc10[v];
    o1[16] = acc11[v];
  }
}

// ---------------------------------------------------------------------------
// Stage 3: pred[m, d] = x_out0[m, :] . W_G[:, d] + b_G[d], m = b*NO + j
// M=2048, K=1024, N=1024.  grid = (16, 128), block = 128 (wave per 16x16 tile)
// ---------------------------------------------------------------------------
__global__ void predgemm_kernel(const float* __restrict__ Xo, const float* __restrict__ WG,
                                const float* __restrict__ bG, float* __restrict__ pred) {
  const int lane = threadIdx.x & 31, half = lane >> 4, lm = lane & 15;
  const int wave = threadIdx.x >> 5;
  const int n0 = (blockIdx.x * 4 + wave) * 16;
  const int m0 = blockIdx.y * 16;
  const int kxo = half ? 2 : 0, kyo = half ? 3 : 1;

  const float* Arow = Xo + (size_t)(m0 + lm) * DO;
  const float* Bcol = WG + n0 + lm;

  v8f acc = {};
  #pragma unroll 4
  for (int k0 = 0; k0 < DO; k0 += 4) {
    const int kx = k0 + kxo, ky = k0 + kyo;
    v2f a, bb;
    a.x = Arow[kx];               a.y = Arow[ky];
    bb.x = Bcol[(size_t)kx * DI]; bb.y = Bcol[(size_t)ky * DI];
    acc = wmma_f32(a, bb, acc);
  }
  const float bias = bG[n0 + lm];
  #pragma unroll
  for (int v = 0; v < 8; ++v) {
    const int m = m0 + v + 8 * half;
    pred[(size_t)m * DI + n0 + lm] = acc[v] + bias;
  }
}

// ---------------------------------------------------------------------------
// Stage 4: S[b, i, j] = (x[b,i,:] . pred[b,j,:]) / sqrt(DI)
// per b: M=1024, N=64, K=1024. grid = (64 Mtiles, 32 b), block = 128 (4 Ntiles)
// ---------------------------------------------------------------------------
__global__ void sgemm_kernel(const float* __restrict__ x, const float* __restrict__ pred,
                             float* __restrict__ S) {
  const int lane = threadIdx.x & 31, half = lane >> 4, lm = lane & 15;
  const int wave = threadIdx.x >> 5;     // N tile: j0 = wave*16
  const int b  = blockIdx.y;
  const int m0 = blockIdx.x * 16;
  const int n0 = wave * 16;
  const int kxo = half ? 2 : 0, kyo = half ? 3 : 1;

  const float* Arow = x    + ((size_t)b * NI + m0 + lm) * DI;
  const float* Brow = pred + ((size_t)b * NO + n0 + lm) * DI;

  v8f acc = {};
  #pragma unroll 4
  for (int k0 = 0; k0 < DI; k0 += 4) {
    const int kx = k0 + kxo, ky = k0 + kyo;
    v2f a, bb;
    a.x = Arow[kx];  a.y = Arow[ky];
    bb.x = Brow[kx]; bb.y = Brow[ky];
    acc = wmma_f32(a, bb, acc);
  }
  const float scale = 0.03125f;  // DI^-0.5 = 1/32
  #pragma unroll
  for (int v = 0; v < 8; ++v) {
    const int i = m0 + v + 8 * half;
    S[((size_t)b * NI + i) * NO + n0 + lm] = acc[v] * scale;
  }
}

// ---------------------------------------------------------------------------
// Stage 5: softmax over j (64) + iter-1 phi, stored transposed phiT[j][i][b]
// phi = f_a * (R*(beta_use + beta_ign) - beta_ign). One thread per (b,i) row.
// ---------------------------------------------------------------------------
__global__ void softmax_phi_kernel(const float* __restrict__ S, const float* __restrict__ fa,
                                   const float* __restrict__ bu, const float* __restrict__ bi,
                                   float* __restrict__ phiT) {
  const int t = blockIdx.x * blockDim.x + threadIdx.x; // t = b*NI + i
  const int b = t >> 10;
  const int i = t & (NI - 1);
  const float* row = S + (size_t)t * NO;
  float mx = -3.0e38f;
  #pragma unroll 8
  for (int j = 0; j < NO; ++j) mx = fmaxf(mx, row[j]);
  float sum = 0.0f;
  #pragma unroll 8
  for (int j = 0; j < NO; ++j) sum += __expf(row[j] - mx);
  const float inv = 1.0f / sum;
  const float f = fa[t];
  for (int j = 0; j < NO; ++j) {
    const float R  = __expf(row[j] - mx) * inv;
    const float bg = bi[i * NO + j];
    const float w  = R * (bu[i * NO + j] + bg) - bg;
    phiT[((size_t)j * NI + i) * B_SZ + b] = f * w;
  }
}

// ---------------------------------------------------------------------------
extern "C" void kernel_launch(void* const* d_in, const int* in_sizes, int n_in,
                              void* d_out, int out_size, void* d_ws, size_t ws_size,
                              hipStream_t stream) {
  (void)in_sizes; (void)n_in; (void)out_size; (void)ws_size;
  const float* x    = (const float*)d_in[0];
  const float* WA   = (const float*)d_in[1];
  const float* bA   = (const float*)d_in[2];
  const float* Wmem = (const float*)d_in[3];
  const float* WG   = (const float*)d_in[4];
  const float* bG   = (const float*)d_in[5];
  const float* bu   = (const float*)d_in[6];
  const float* bi   = (const float*)d_in[7];
  float* out = (float*)d_out;

  float* ws   = (float*)d_ws;
  float* fa   = ws;                                  // 32768 floats
  float* phiT = fa   + (size_t)B_SZ * NI;            // 2M floats [j][i][b]
  float* xo0  = phiT + (size_t)NO * NI * B_SZ;       // 2M floats [b][j][h]
  float* pred = xo0  + (size_t)B_SZ * NO * DO;       // 2M floats [b][j][d]
  float* S    = pred + (size_t)B_SZ * NO * DI;       // 2M floats [b][i][j]

  const size_t lds_bytes = (size_t)NI * B_SZ * sizeof(float);  // 128 KB

  fa_kernel<<<dim3((B_SZ * NI) / 8), dim3(256), 0, stream>>>(x, WA, bA, fa);
  phi0_kernel<<<dim3((NO * NI * B_SZ) / 256), dim3(256), 0, stream>>>(fa, bu, bi, phiT);
  memgemm_kernel<<<dim3(4, 64), dim3(256), lds_bytes, stream>>>(phiT, Wmem, xo0);
  predgemm_kernel<<<dim3(16, 128), dim3(128), 0, stream>>>(xo0, WG, bG, pred);
  sgemm_kernel<<<dim3(64, 32), dim3(128), 0, stream>>>(x, pred, S);
  softmax_phi_kernel<<<dim3((B_SZ * NI) / 256), dim3(256), 0, stream>>>(S, fa, bu, bi, phiT);
  memgemm_kernel<<<dim3(4, 64), dim3(256), lds_bytes, stream>>>(phiT, Wmem, out);
}